// TTLinear_89893665505530
// MI455X (gfx1250) — compile-verified
//
#include <hip/hip_runtime.h>

// TT-linear (ranks 1,4,4,4,1; modes 8^4; 4096->4096) fused per-row in LDS.
// Every stage is a dense GEMM  S'(MxN) = S(MxK) * W(KxN)  with M=512/row,
// K in {8,32,32,32}, N in {32,32,32,8}, W = core reshaped (r0*8)x(8*r1).
// Computed in exact f32 via V_WMMA_F32_16X16X4_F32.

typedef __attribute__((ext_vector_type(2))) float v2f;
typedef __attribute__((ext_vector_type(8))) float v8f;

#define TT_IN_DIM 4096
#define TT_OUT_DIM 4096
#define TT_BLOCK 256                  // 8 wave32 waves, fixed
#define TT_NWAVES (TT_BLOCK / 32)
#define SBUF 16384                    // floats per ping-pong LDS buffer
#define LDS_FLOATS (2 * SBUF + 2560)  // + all 4 cores (256+1024+1024+256)
#define LDS_BYTES (LDS_FLOATS * 4)

// One TT stage. Input layout (q, o, i, p), output layout (s, o, c, p).
//   idx_in  = q*(OSZ*8*PSZ) + o*(8*PSZ) + i*PSZ + p
//   idx_out = s*(OSZ*8*PSZ) + o*(8*PSZ) + c*PSZ + p
// GEMM view: A[m=(o,p), kk=(q,i)],  W[kk, n=(c,s)],  D[m, n].
// All index math is unsigned so /,% by power-of-two lower to shift/and.
template <unsigned OSZ, unsigned PSZ, unsigned R0, unsigned R1>
__device__ __forceinline__ void tt_stage(const float* __restrict__ Sin,
                                         float* __restrict__ Sout,
                                         const float* __restrict__ W) {
  constexpr unsigned K = R0 * 8;
  constexpr unsigned N = 8 * R1;
  constexpr unsigned KSTEPS = K / 4;       // wmma_f32 is 16x16x4
  constexpr unsigned NT = (N + 15) / 16;   // N tiles (stage 3 pads 8 -> 16)
  constexpr unsigned MT = 512 / 16;        // 32 M tiles per row
  constexpr unsigned TILES_PER_WAVE = MT / TT_NWAVES;  // exactly 4, uniform
  constexpr unsigned QS = OSZ * 8 * PSZ;   // q / s stride

  const unsigned lane = threadIdx.x & 31u;
  // Wave id is uniform across the wave; tell the compiler so loop control and
  // tile base addresses scalarize, and so WMMAs sit under uniform EXEC.
  const unsigned wave =
      (unsigned)__builtin_amdgcn_readfirstlane(threadIdx.x >> 5);
  const unsigned lhalf = lane >> 4;        // 0: lanes 0-15, 1: lanes 16-31
  const unsigned l16 = lane & 15u;

#pragma unroll
  for (unsigned nt = 0; nt < NT; ++nt) {
    const unsigned n = nt * 16 + l16;
    const bool nvalid = (n < N);

    // Hoist B fragments for this N tile (shared by all M tiles).
    // f32 B 4x16 layout: v0 holds K=kbase+2*lhalf, v1 holds K=kbase+2*lhalf+1.
    v2f bfrag[KSTEPS];
#pragma unroll
    for (unsigned ks = 0; ks < KSTEPS; ++ks) {
      const unsigned kk = ks * 4 + lhalf * 2;
      bfrag[ks].x = nvalid ? W[kk * N + n] : 0.0f;
      bfrag[ks].y = nvalid ? W[(kk + 1) * N + n] : 0.0f;
    }
    const unsigned cc = nvalid ? (n / R1) : 0u;
    const unsigned ss = nvalid ? (n % R1) : 0u;
    const unsigned ncol_off = ss * QS + cc * PSZ;

#pragma unroll
    for (unsigned t = 0; t < TILES_PER_WAVE; ++t) {
      const unsigned mt = wave + t * TT_NWAVES;

      // A fragment addressing: lane l16 supplies row m = mt*16 + l16.
      const unsigned mA = mt * 16 + l16;
      const unsigned oA = mA / PSZ;        // power of two -> shift
      const unsigned pA = mA % PSZ;        // power of two -> and
      const unsigned rowA = oA * (8 * PSZ) + pA;

      v8f acc = {};
#pragma unroll
      for (unsigned ks = 0; ks < KSTEPS; ++ks) {
        const unsigned kk = ks * 4 + lhalf * 2;  // even -> kk,kk+1 share q
        const unsigned q = kk >> 3;
        const unsigned i = kk & 7u;
        const unsigned aoff = q * QS + rowA + i * PSZ;
        v2f a;
        a.x = Sin[aoff];
        a.y = Sin[aoff + PSZ];             // (q, i+1)
        acc = __builtin_amdgcn_wmma_f32_16x16x4_f32(
            false, a, false, bfrag[ks], (short)0, acc, false, false);
      }

      // Scatter D (lane n fixed; VGPR v -> M = v + 8*lhalf) into (s,o,c,p).
      if (nvalid) {
#pragma unroll
        for (unsigned v = 0; v < 8; ++v) {
          const unsigned mD = mt * 16 + lhalf * 8 + v;
          const unsigned oD = mD / PSZ;
          const unsigned pD = mD % PSZ;
          Sout[ncol_off + oD * (8 * PSZ) + pD] = acc[v];
        }
      }
    }
  }
}

__global__ __launch_bounds__(TT_BLOCK) void tt_linear_kernel(
    const float* __restrict__ x, const float* __restrict__ c0,
    const float* __restrict__ c1, const float* __restrict__ c2,
    const float* __restrict__ c3, const float* __restrict__ bias,
    float* __restrict__ out) {
  extern __shared__ float smem[];
  float* bufA = smem;              // 16384 floats
  float* bufB = smem + SBUF;       // 16384 floats
  float* W = smem + 2 * SBUF;      // cores: [0,256) [256,1280) [1280,2304) [2304,2560)

  const unsigned tid = threadIdx.x;
  const unsigned long long row = blockIdx.x;

  // Stage cores into LDS (tiny, reused ~1400x per WG).
  for (unsigned i = tid; i < 256; i += TT_BLOCK) W[i] = c0[i];
  for (unsigned i = tid; i < 1024; i += TT_BLOCK) W[256 + i] = c1[i];
  for (unsigned i = tid; i < 1024; i += TT_BLOCK) W[1280 + i] = c2[i];
  for (unsigned i = tid; i < 256; i += TT_BLOCK) W[2304 + i] = c3[i];

  // Stage this row of x into bufA (coalesced b128 loads).
  const float4* xr = (const float4*)(x + row * TT_IN_DIM);
  float4* bA4 = (float4*)bufA;
#pragma unroll
  for (unsigned i = 0; i < TT_IN_DIM / 4 / TT_BLOCK; ++i)
    bA4[tid + i * TT_BLOCK] = xr[tid + i * TT_BLOCK];
  __syncthreads();

  // 4 fused stage-GEMMs, ping-ponging LDS buffers.
  tt_stage<1, 512, 1, 4>(bufA, bufB, W);          // 4096  -> 16384
  __syncthreads();
  tt_stage<8, 64, 4, 4>(bufB, bufA, W + 256);     // 16384 -> 16384
  __syncthreads();
  tt_stage<64, 8, 4, 4>(bufA, bufB, W + 1280);    // 16384 -> 16384
  __syncthreads();
  tt_stage<512, 1, 4, 1>(bufB, bufA, W + 2304);   // 16384 -> 4096
  __syncthreads();

  // bufA now holds the 4096 output features in order (o*8 + c).
  float* orow = out + row * TT_OUT_DIM;
#pragma unroll
  for (unsigned i = 0; i < TT_OUT_DIM / TT_BLOCK; ++i)
    orow[tid + i * TT_BLOCK] =
        bufA[tid + i * TT_BLOCK] + bias[tid + i * TT_BLOCK];
}

extern "C" void kernel_launch(void* const* d_in, const int* in_sizes, int n_in,
                              void* d_out, int out_size, void* d_ws,
                              size_t ws_size, hipStream_t stream) {
  (void)in_sizes; (void)n_in; (void)d_ws; (void)ws_size; (void)out_size;
  const float* x = (const float*)d_in[0];
  const float* c0 = (const float*)d_in[1];
  const float* c1 = (const float*)d_in[2];
  const float* c2 = (const float*)d_in[3];
  const float* c3 = (const float*)d_in[4];
  const float* bias = (const float*)d_in[5];
  float* out = (float*)d_out;

  // Opt in to >64KB dynamic LDS (gfx1250 WGP has 320KB).
  (void)hipFuncSetAttribute((const void*)tt_linear_kernel,
                            hipFuncAttributeMaxDynamicSharedMemorySize,
                            LDS_BYTES);

  const int rows = 4 * 2048;  // (4, 2048, 4096) -> 8192 rows
  tt_linear_kernel<<<rows, TT_BLOCK, LDS_BYTES, stream>>>(x, c0, c1, c2, c3,
                                                          bias, out);
}